// GraphSAGE_18004502905473
// MI455X (gfx1250) — compile-verified
//
#include <hip/hip_runtime.h>
#include <math.h>

#define NNODES 50000
#define CH     128        // in/out channels
#define KTOT   256        // concatenated K = [agg | x]
#define WSTR   129        // wlds row stride (k-major, +1 pad -> conflict-free)
#define ASTR   260        // alds row stride (+4 pad -> disjoint bank halves for b64)

typedef __attribute__((ext_vector_type(2))) float v2f;
typedef __attribute__((ext_vector_type(8))) float v8f;

// ---------------- scatter: mean-aggregation numerator + counts ----------------
__global__ __launch_bounds__(256)
void sage_scatter(const float* __restrict__ x, const long long* __restrict__ ei,
                  float* __restrict__ summed, float* __restrict__ counts, int E) {
    int g    = blockIdx.x * blockDim.x + threadIdx.x;
    int e    = g >> 5;
    int lane = g & 31;
    if (e >= E) return;
    long long src = ei[e];
    long long dst = ei[(long long)E + e];
    const float4 v = *(const float4*)(x + (size_t)src * CH + lane * 4);
    float* d = summed + (size_t)dst * CH + lane * 4;
    __hip_atomic_fetch_add(d + 0, v.x, __ATOMIC_RELAXED, __HIP_MEMORY_SCOPE_AGENT);
    __hip_atomic_fetch_add(d + 1, v.y, __ATOMIC_RELAXED, __HIP_MEMORY_SCOPE_AGENT);
    __hip_atomic_fetch_add(d + 2, v.z, __ATOMIC_RELAXED, __HIP_MEMORY_SCOPE_AGENT);
    __hip_atomic_fetch_add(d + 3, v.w, __ATOMIC_RELAXED, __HIP_MEMORY_SCOPE_AGENT);
    if (lane == 0)
        __hip_atomic_fetch_add(counts + dst, 1.0f, __ATOMIC_RELAXED, __HIP_MEMORY_SCOPE_AGENT);
}

// -------- fused: mean + [agg|x] @ [Wl;Wr]^T + b + log_softmax (f32 WMMA) --------
__global__ __launch_bounds__(128)
void sage_fused(const float* __restrict__ x,  const float* __restrict__ Wl,
                const float* __restrict__ Wr, const float* __restrict__ bias,
                const float* __restrict__ summed, const float* __restrict__ counts,
                float* __restrict__ out) {
    extern __shared__ float smem[];
    float* wlds = smem;                       // [KTOT][WSTR] : wlds[k*WSTR+n] = Wc[n][k]
    float* alds = smem + KTOT * WSTR;         // per-wave [32][ASTR]

    const int tid  = threadIdx.x;
    const int lane = tid & 31;
    const int wave = tid >> 5;
    const int ln   = lane & 15;               // N / M position within half
    const int hi   = lane >> 4;               // lane half selects K pair (0,1)/(2,3)
    const int koff = hi * 2;

    // ---- stage concatenated weights, k-major (coalesced global reads) ----
    for (int i = tid; i < CH * KTOT; i += 128) {
        int n = i >> 8;            // i / 256
        int k = i & 255;
        float v = (k < CH) ? Wl[n * CH + k] : Wr[n * CH + (k - CH)];
        wlds[k * WSTR + n] = v;
    }

    // ---- stage this wave's 32 A-rows: [agg(row) | x(row)] ----
    const int rowBase = blockIdx.x * 128 + wave * 32;
    float* aw = alds + wave * 32 * ASTR;
    for (int r = 0; r < 32; ++r) {
        int row = rowBase + r; if (row >= NNODES) row = NNODES - 1;
        float inv = 1.0f / fmaxf(counts[row], 1.0f);
        float4 s  = *(const float4*)(summed + (size_t)row * CH + lane * 4);
        float4 xv = *(const float4*)(x      + (size_t)row * CH + lane * 4);
        float* dp = aw + r * ASTR + lane * 4;
        dp[0] = s.x * inv; dp[1] = s.y * inv; dp[2] = s.z * inv; dp[3] = s.w * inv;
        dp[CH + 0] = xv.x; dp[CH + 1] = xv.y; dp[CH + 2] = xv.z; dp[CH + 3] = xv.w;
    }
    __syncthreads();

    // ---- K-loop: D = A(32x256) x B(256x128), f32 WMMA 16x16x4 ----
    v8f acc[2][8];
    const v8f z = {0.f,0.f,0.f,0.f,0.f,0.f,0.f,0.f};
    for (int mt = 0; mt < 2; ++mt)
        for (int nt = 0; nt < 8; ++nt) acc[mt][nt] = z;

    for (int k0 = 0; k0 < KTOT; k0 += 4) {
        const int kb = k0 + koff;             // VGPR0 -> K=kb, VGPR1 -> K=kb+1
        v2f b[8];
        #pragma unroll
        for (int nt = 0; nt < 8; ++nt) {
            int n = nt * 16 + ln;
            b[nt][0] = wlds[(kb + 0) * WSTR + n];
            b[nt][1] = wlds[(kb + 1) * WSTR + n];
        }
        #pragma unroll
        for (int mt = 0; mt < 2; ++mt) {
            const float* ap = aw + (mt * 16 + ln) * ASTR + kb;
            v2f a; a[0] = ap[0]; a[1] = ap[1];
            #pragma unroll
            for (int nt = 0; nt < 8; ++nt)
                acc[mt][nt] = __builtin_amdgcn_wmma_f32_16x16x4_f32(
                    false, a, false, b[nt], (short)0, acc[mt][nt], false, false);
        }
    }

    // ---- bias + row-wise log_softmax + store ----
    float bv[8];
    #pragma unroll
    for (int nt = 0; nt < 8; ++nt) bv[nt] = bias[nt * 16 + ln];

    for (int mt = 0; mt < 2; ++mt) {
        #pragma unroll
        for (int v = 0; v < 8; ++v) {
            // this VGPR slot holds rows (v) for lanes 0-15 and (v+8) for lanes 16-31
            float t[8]; float mx = -3.0e38f;
            #pragma unroll
            for (int nt = 0; nt < 8; ++nt) { t[nt] = acc[mt][nt][v] + bv[nt]; mx = fmaxf(mx, t[nt]); }
            #pragma unroll
            for (int off = 1; off < 16; off <<= 1) mx = fmaxf(mx, __shfl_xor(mx, off));
            float sum = 0.f;
            #pragma unroll
            for (int nt = 0; nt < 8; ++nt) sum += expf(t[nt] - mx);
            #pragma unroll
            for (int off = 1; off < 16; off <<= 1) sum += __shfl_xor(sum, off);
            const float lse = mx + logf(sum);
            const int row = rowBase + mt * 16 + v + hi * 8;
            if (row < NNODES) {
                #pragma unroll
                for (int nt = 0; nt < 8; ++nt)
                    out[(size_t)row * CH + nt * 16 + ln] = t[nt] - lse;
            }
        }
    }
}

extern "C" void kernel_launch(void* const* d_in, const int* in_sizes, int n_in,
                              void* d_out, int out_size, void* d_ws, size_t ws_size,
                              hipStream_t stream) {
    const float*     x  = (const float*)d_in[0];
    const long long* ei = (const long long*)d_in[1];
    const float*     Wl = (const float*)d_in[2];
    const float*     Wr = (const float*)d_in[3];
    const float*     b  = (const float*)d_in[4];
    float*           out = (float*)d_out;

    const int E = in_sizes[1] / 2;

    float* summed = (float*)d_ws;
    float* counts = summed + (size_t)NNODES * CH;

    hipMemsetAsync(d_ws, 0, ((size_t)NNODES * CH + NNODES) * sizeof(float), stream);

    // one wave per edge
    const long long threadsTotal = (long long)E * 32;
    const int sblocks = (int)((threadsTotal + 255) / 256);
    sage_scatter<<<sblocks, 256, 0, stream>>>(x, ei, summed, counts, E);

    const size_t lds = (size_t)(KTOT * WSTR + 4 * 32 * ASTR) * sizeof(float);
    const int fblocks = (NNODES + 127) / 128;
    sage_fused<<<fblocks, 128, lds, stream>>>(x, Wl, Wr, b, summed, counts, out);
}